// PrototypeSuppressor_11733850653128
// MI455X (gfx1250) — compile-verified
//
#include <hip/hip_runtime.h>
#include <stdint.h>

typedef __attribute__((ext_vector_type(2))) float v2f;
typedef __attribute__((ext_vector_type(8))) float v8f;

#define NB     4        // batch
#define DDIM   1024     // hidden dim
#define SLEN   1024     // seq positions pooled for emb
#define NPROTO 64
#define VOCAB  32000
#define KTOP   20

// Order-preserving float -> uint mapping (larger float <=> larger uint)
__device__ __forceinline__ unsigned mapf(float f) {
    unsigned u = __float_as_uint(f);
    return u ^ ((unsigned)((int)u >> 31) | 0x80000000u);
}

// ---------------------------------------------------------------------------
// Kernel 0: zero padding rows 4..15 of the 16x1024 emb buffer so WMMA A-tile
// loads are unconditional and contribute exact zeros.
// ---------------------------------------------------------------------------
__global__ void zero_pad_kernel(float* __restrict__ emb) {
    int idx = blockIdx.x * blockDim.x + threadIdx.x;   // 0 .. 12287
    emb[NB * DDIM + idx] = 0.0f;
}

// ---------------------------------------------------------------------------
// Kernel 1: emb[b][d] = mean over s of hidden_states[b][s][d]   (coalesced in d)
// grid: 16 blocks (b*4 + dchunk), 256 threads
// ---------------------------------------------------------------------------
__global__ void mean_kernel(const float* __restrict__ hs, float* __restrict__ emb) {
    int b = blockIdx.x >> 2;
    int d = ((blockIdx.x & 3) << 8) + threadIdx.x;
    const float* p = hs + (size_t)b * SLEN * DDIM + d;
    float sum = 0.0f;
#pragma unroll 8
    for (int s = 0; s < SLEN; ++s) sum += p[(size_t)s * DDIM];
    emb[b * DDIM + d] = sum * (1.0f / (float)SLEN);
}

// ---------------------------------------------------------------------------
// Kernel 2: inverse norms + WMMA f32 GEMM  C[16x64] = emb_n @ pro_n^T,
// row-mean over the 64 prototypes, flags[b] = (sim > 0).
// 1 block, 128 threads = 4 waves; wave w owns N-tile [16w, 16w+16).
// Uses V_WMMA_F32_16X16X4_F32 (full fp32 precision for the sign decision).
// ---------------------------------------------------------------------------
__global__ void simflag_kernel(const float* __restrict__ prototypes,
                               const float* __restrict__ emb,
                               int* __restrict__ flags) {
    __shared__ float inv_emb[4];
    __shared__ float inv_pro[NPROTO];
    __shared__ float simPartial[4];

    const int tid  = threadIdx.x;
    const int wave = tid >> 5;
    const int lane = tid & 31;

    // --- inverse norms of emb rows (wave w -> row w) ---
    if (wave < 4) {
        float ss = 0.0f;
        const float* r = emb + wave * DDIM;
        for (int i = lane; i < DDIM; i += 32) { float v = r[i]; ss += v * v; }
        for (int m = 16; m >= 1; m >>= 1) ss += __shfl_xor(ss, m, 32);
        if (lane == 0) inv_emb[wave] = 1.0f / fmaxf(sqrtf(ss), 1e-8f);
    }
    // --- inverse norms of prototype rows ---
    for (int row = wave; row < NPROTO; row += 4) {
        float ss = 0.0f;
        const float* r = prototypes + row * DDIM;
        for (int i = lane; i < DDIM; i += 32) { float v = r[i]; ss += v * v; }
        for (int m = 16; m >= 1; m >>= 1) ss += __shfl_xor(ss, m, 32);
        if (lane == 0) inv_pro[row] = 1.0f / fmaxf(sqrtf(ss), 1e-8f);
    }
    if (tid < 4) simPartial[tid] = 0.0f;
    __syncthreads();

    // --- WMMA GEMM: A = emb_n (16x1024, rows 4..15 are zero), B = pro_n^T ---
    // A 16x4 f32 layout: lanes 0-15 hold M=lane, {v0,v1} = K{0,1}; lanes 16-31 K{2,3}
    // B 4x16 mirrors:    lanes 0-15 hold N=lane, {v0,v1} = K{0,1}; lanes 16-31 K{2,3}
    const int mrow = lane & 15;
    const int koff = (lane < 16) ? 0 : 2;
    const int nrow = (wave << 4) + mrow;                 // prototype row
    const float ia = (mrow < 4) ? inv_emb[mrow] : 0.0f;  // zero pad rows
    const float ib = inv_pro[nrow];
    const float* arow = emb + mrow * DDIM;
    const float* brow = prototypes + nrow * DDIM;

    v8f c = {};
    for (int k = 0; k < DDIM; k += 4) {
        float2 af = *(const float2*)(arow + k + koff);   // 8B aligned (k+koff even)
        float2 bf = *(const float2*)(brow + k + koff);
        v2f a; a.x = af.x * ia; a.y = af.y * ia;
        v2f b; b.x = bf.x * ib; b.y = bf.y * ib;
        c = __builtin_amdgcn_wmma_f32_16x16x4_f32(
                /*neg_a=*/false, a, /*neg_b=*/false, b,
                /*c_mod=*/(short)0, c, /*reuse_a=*/false, /*reuse_b=*/false);
    }

    // C layout: VGPR v, lanes 0-15 -> M=v, N=16w+lane. Sum rows 0..3 over N.
    for (int m = 0; m < 4; ++m) {
        float v = (lane < 16) ? c[m] : 0.0f;
        for (int msk = 16; msk >= 1; msk >>= 1) v += __shfl_xor(v, msk, 32);
        if (lane == 0) atomicAdd(&simPartial[m], v);
    }
    __syncthreads();
    if (tid < 4)
        flags[tid] = (simPartial[tid] * (1.0f / (float)NPROTO) > 0.0f) ? 1 : 0;
}

// ---------------------------------------------------------------------------
// Kernel 3: per-row radix-select of the 20th largest logit + suppression.
// One 1024-thread block per row (4096 rows). 4 MSD radix passes (8-bit digits)
// on the order-preserving uint key; repeated row reads hit L2 (row is 128 KB).
// ---------------------------------------------------------------------------
__global__ void __launch_bounds__(1024)
topk_suppress_kernel(const float* __restrict__ logits,
                     const int* __restrict__ flags,
                     float* __restrict__ out) {
    const size_t row = blockIdx.x;
    const int tid = threadIdx.x;
    const int b = (int)(row >> 10);
    const float4* in4 = (const float4*)(logits + row * VOCAB);
    float4* out4 = (float4*)(out + row * VOCAB);
    const int n4 = VOCAB / 4;                            // 8000

    if (!flags[b]) {                                      // plain stream copy
        for (int i = tid; i < n4; i += 1024) out4[i] = in4[i];
        return;
    }

    __shared__ unsigned hist[256];
    __shared__ unsigned s_prefix;
    __shared__ int s_k;
    if (tid == 0) { s_prefix = 0u; s_k = KTOP; }

    for (int pass = 0; pass < 4; ++pass) {
        const int shift = 24 - (pass << 3);
        if (tid < 256) hist[tid] = 0u;
        __syncthreads();
        const unsigned prefix = s_prefix;
        const unsigned maskHigh = (pass == 0) ? 0u : (0xFFFFFFFFu << (shift + 8));

        for (int i = tid; i < n4; i += 1024) {
            float4 f = in4[i];
            unsigned u0 = mapf(f.x), u1 = mapf(f.y), u2 = mapf(f.z), u3 = mapf(f.w);
            if ((u0 & maskHigh) == prefix) atomicAdd(&hist[(u0 >> shift) & 255u], 1u);
            if ((u1 & maskHigh) == prefix) atomicAdd(&hist[(u1 >> shift) & 255u], 1u);
            if ((u2 & maskHigh) == prefix) atomicAdd(&hist[(u2 >> shift) & 255u], 1u);
            if ((u3 & maskHigh) == prefix) atomicAdd(&hist[(u3 >> shift) & 255u], 1u);
        }
        __syncthreads();
        if (tid == 0) {
            int k = s_k; unsigned cum = 0; int dsel = 0;
            for (int d = 255; d >= 0; --d) {
                unsigned cnt = hist[d];
                if (cum + cnt >= (unsigned)k) { dsel = d; break; }
                cum += cnt;
            }
            s_prefix = prefix | ((unsigned)dsel << shift);
            s_k = k - (int)cum;
        }
        __syncthreads();
    }

    const unsigned thr = s_prefix;   // key of the 20th-largest element
    for (int i = tid; i < n4; i += 1024) {
        float4 f = in4[i];
        f.x = (mapf(f.x) >= thr) ? -100.0f : f.x;
        f.y = (mapf(f.y) >= thr) ? -100.0f : f.y;
        f.z = (mapf(f.z) >= thr) ? -100.0f : f.z;
        f.w = (mapf(f.w) >= thr) ? -100.0f : f.w;
        out4[i] = f;
    }
}

// ---------------------------------------------------------------------------
extern "C" void kernel_launch(void* const* d_in, const int* in_sizes, int n_in,
                              void* d_out, int out_size, void* d_ws, size_t ws_size,
                              hipStream_t stream) {
    const float* prototypes = (const float*)d_in[0];   // [64, 1024]
    const float* hidden     = (const float*)d_in[1];   // [4, 1024, 1024]
    const float* logits     = (const float*)d_in[2];   // [4, 1, 1024, 32000]
    float* out = (float*)d_out;

    float* emb = (float*)d_ws;                               // 16 x 1024 f32 (rows 4..15 zero)
    int* flags = (int*)((char*)d_ws + 16 * DDIM * sizeof(float));

    zero_pad_kernel<<<12, 1024, 0, stream>>>(emb);
    mean_kernel<<<16, 256, 0, stream>>>(hidden, emb);
    simflag_kernel<<<1, 128, 0, stream>>>(prototypes, emb, flags);
    topk_suppress_kernel<<<NB * SLEN, 1024, 0, stream>>>(logits, flags, out);
}